// BiaffineLabeler_45887430590981
// MI455X (gfx1250) — compile-verified
//
#include <hip/hip_runtime.h>

typedef __attribute__((ext_vector_type(2))) float v2f;
typedef __attribute__((ext_vector_type(4))) float v4f;
typedef __attribute__((ext_vector_type(8))) float v8f;

#define BD   8
#define TT   256
#define DD   1024
#define DL   512
#define NLAB 50
#define ROWS (BD * TT)   // 2048 token rows

// ---------------------------------------------------------------------------
// gfx1250 async global->LDS copy (16B per lane), tracked by ASYNCcnt.
// LDS byte address = low 32 bits of the flat pointer (ISA: LDS aperture maps
// addr[31:0] directly to the wave's LDS offset).
// ---------------------------------------------------------------------------
static __device__ __forceinline__ void async_ld_b128(const void* lds_dst,
                                                     const void* gsrc) {
  unsigned l = (unsigned)(unsigned long long)lds_dst;
  asm volatile("global_load_async_to_lds_b128 %0, %1, off"
               :: "v"(l), "v"(gsrc)
               : "memory");
}

// ---------------------------------------------------------------------------
// Stage 1: label projections.
//   out[row, j] = sum_k in[row,k] * Wm[k,j] + bv[j]
// grid = (128 row-tiles, 4 j-chunks of 128, 2 sources {dep, head}), block 128.
// ---------------------------------------------------------------------------
__global__ void proj_label_kernel(const float* __restrict__ dep,
                                  const float* __restrict__ head,
                                  const float* __restrict__ Wdep,
                                  const float* __restrict__ bdep,
                                  const float* __restrict__ Whead,
                                  const float* __restrict__ bhead,
                                  float* __restrict__ ws) {
  extern __shared__ float lds[];                // AT[k][t] : 1024*16 floats
  const int rt  = blockIdx.x;
  const int jc  = blockIdx.y;
  const int src = blockIdx.z;
  const float* in = src ? head  : dep;
  const float* Wm = src ? Whead : Wdep;
  const float* bv = src ? bhead : bdep;
  float* out = ws + (size_t)src * ((size_t)ROWS * DL);

  const int tid = threadIdx.x;

  for (int idx = tid; idx < 16 * (DD / 4); idx += 128) {
    const int t  = idx >> 8;
    const int k4 = idx & 255;
    v4f v = *(const v4f*)(in + (size_t)(rt * 16 + t) * DD + k4 * 4);
#pragma unroll
    for (int i = 0; i < 4; ++i) lds[(k4 * 4 + i) * 16 + t] = v[i];
  }
  __syncthreads();

  const int lane = tid & 31;
  const int wave = tid >> 5;
  const int r    = lane & 15;
  const int hi   = lane >> 4;
  const int j0   = jc * 128 + wave * 32;

  v8f c0 = {};
  v8f c1 = {};
  for (int k = 0; k < DD; k += 4) {
    v2f a;
    a.x = lds[(k + 2 * hi) * 16 + r];
    a.y = lds[(k + 2 * hi + 1) * 16 + r];
    const float* wp = Wm + (size_t)(k + 2 * hi) * DL + j0 + r;
    v2f b0, b1;
    b0.x = wp[0];       b0.y = wp[DL];
    b1.x = wp[16];      b1.y = wp[DL + 16];
    c0 = __builtin_amdgcn_wmma_f32_16x16x4_f32(false, a, false, b0, (short)0, c0, false, false);
    c1 = __builtin_amdgcn_wmma_f32_16x16x4_f32(false, a, false, b1, (short)0, c1, false, false);
  }

  const float bias0 = bv[j0 + r];
  const float bias1 = bv[j0 + 16 + r];
#pragma unroll
  for (int v = 0; v < 8; ++v) {
    const int row = rt * 16 + v + 8 * hi;
    out[(size_t)row * DL + j0 + r]      = c0[v] + bias0;
    out[(size_t)row * DL + j0 + 16 + r] = c1[v] + bias1;
  }
}

// ---------------------------------------------------------------------------
// Stage 2: biaffine scores.
//   Yt[d,t] = sum_e W[n,d,e] * sel[t,e]      (WMMA: A = W[n], B = sel^T)
//   logits[t,n] = sum_d dep[t,d] * Yt[d,t] + bias[n]
// grid = (32 supertiles of 64 tokens, 50 labels), block 256 (8 waves).
// Each workgroup amortizes the staged W[n] over 4 token-tiles (4x less L2
// traffic). W streamed in 512x32 e-chunks via global_load_async_to_lds_b128,
// double-buffered (ASYNCcnt<=16): chunk c+1 flies while chunk c feeds WMMAs.
// Wc rows stride 36 floats (16B-aligned rows, conflict-free b64 A reads);
// sel tiles transposed, stride 16 (conflict-free B reads). dep tiles are
// staged AFTER the chunk loop into the dead W buffer (LDS alias) to stay
// within the 320KB WGP budget: 128KB selT + 144KB Wc + red.
// ---------------------------------------------------------------------------
#define WCH   32                    // e-columns per chunk
#define WPAD  36                    // Wc row stride in floats (32 + 4)
#define WBUF  (DL * WPAD)           // one W chunk buffer: 18432 floats
#define NCH   (DL / WCH)            // 16 chunks
#define TPT   (DL * 16)             // one transposed 16-token tile: 8192 floats

__global__ void biaffine_kernel(const float* __restrict__ ws,
                                const int* __restrict__ hidx,
                                const float* __restrict__ W,
                                const float* __restrict__ bias,
                                float* __restrict__ out) {
  extern __shared__ float lds[];
  float* selT = lds;                       // 4 tiles * 512*16 = 32768 floats
  float* Wc   = lds + 4 * TPT;             // 2 * WBUF (double buffer)
  float* red  = Wc + 2 * WBUF;             // 256
  float* depTa = Wc;                       // alias: dep tiles after main loop

  const float* dep_label  = ws;
  const float* head_label = ws + (size_t)ROWS * DL;

  const int tt  = blockIdx.x;              // 0..31 supertiles of 64 tokens
  const int n   = blockIdx.y;
  const int b   = tt >> 2;
  const int t0  = (tt & 3) * 64;
  const int tid = threadIdx.x;

  const float* Wn = W + (size_t)n * DL * DL;

  // Issue chunk 0 asynchronously before staging the sel tiles.
  // Segment s: d = s>>3 (row), e4 = s&7 (16B piece) -> 8 lanes cover one 128B
  // row segment (coalesced); 16 async instructions per wave per chunk.
#pragma unroll
  for (int i = 0; i < 16; ++i) {
    const int s = tid + i * 256;
    const int d = s >> 3, e4 = s & 7;
    async_ld_b128(Wc + d * WPAD + e4 * 4, Wn + (size_t)d * DL + e4 * 4);
  }

  // Gather + transpose 4 selected-head tiles (64 tokens x 512) into LDS.
  for (int idx = tid; idx < 64 * (DL / 4); idx += 256) {
    const int t  = idx >> 7;               // 0..63
    const int e4 = idx & 127;
    const int hrow = b * TT + hidx[b * TT + t0 + t];
    v4f sv = *(const v4f*)(head_label + (size_t)hrow * DL + e4 * 4);
    float* dst = selT + (t >> 4) * TPT + (t & 15);
#pragma unroll
    for (int i = 0; i < 4; ++i) dst[(e4 * 4 + i) * 16] = sv[i];
  }

  const int lane = tid & 31;
  const int wave = tid >> 5;
  const int r    = lane & 15;
  const int hi   = lane >> 4;
  const int dwv  = wave * 64;              // this wave's d-range base

  v8f acc[16] = {};                        // [tile][mt] accumulators

  for (int c = 0; c < NCH; ++c) {
    const int e0 = c * WCH;
    if (c + 1 < NCH) {
      float* nb = Wc + ((c + 1) & 1) * WBUF;
      const float* gs = Wn + (e0 + WCH);
#pragma unroll
      for (int i = 0; i < 16; ++i) {
        const int s = tid + i * 256;
        const int d = s >> 3, e4 = s & 7;
        async_ld_b128(nb + d * WPAD + e4 * 4, gs + (size_t)d * DL + e4 * 4);
      }
      asm volatile("s_wait_asynccnt 16" ::: "memory");  // retire chunk c only
    } else {
      asm volatile("s_wait_asynccnt 0" ::: "memory");
    }
    __syncthreads();   // chunk c visible to all waves

    const float* Wb = Wc + (c & 1) * WBUF;
#pragma unroll
    for (int k = 0; k < WCH; k += 4) {
      // A operands: 4 M-tiles of W[n], reused across all 4 token tiles.
      v2f a[4];
#pragma unroll
      for (int mt = 0; mt < 4; ++mt)
        a[mt] = *(const v2f*)(Wb + (dwv + mt * 16 + r) * WPAD + (k + 2 * hi));
#pragma unroll
      for (int tl = 0; tl < 4; ++tl) {
        v2f bvv;
        bvv.x = selT[tl * TPT + (e0 + k + 2 * hi) * 16 + r];
        bvv.y = selT[tl * TPT + (e0 + k + 2 * hi + 1) * 16 + r];
#pragma unroll
        for (int mt = 0; mt < 4; ++mt)
          acc[tl * 4 + mt] = __builtin_amdgcn_wmma_f32_16x16x4_f32(
              false, a[mt], false, bvv, (short)0, acc[tl * 4 + mt],
              false, false);
      }
    }
    __syncthreads();   // all waves done reading buffer (c&1) before reuse
  }

  // Stage dep tiles into the now-dead W buffer region (alias).
  for (int idx = tid; idx < 64 * (DL / 4); idx += 256) {
    const int t  = idx >> 7;
    const int e4 = idx & 127;
    const int row = b * TT + t0 + t;
    v4f dv = *(const v4f*)(dep_label + (size_t)row * DL + e4 * 4);
    float* dst = depTa + (t >> 4) * TPT + (t & 15);
#pragma unroll
    for (int i = 0; i < 4; ++i) dst[(e4 * 4 + i) * 16] = dv[i];
  }
  __syncthreads();

  // logits[t] = sum_d dep[t,d] * Yt[d,t] + bias[n], one token tile at a time.
  const float bn = bias[n];
  for (int tl = 0; tl < 4; ++tl) {
    float partial = 0.f;
#pragma unroll
    for (int mt = 0; mt < 4; ++mt) {
      const int d0 = dwv + mt * 16;
#pragma unroll
      for (int v = 0; v < 8; ++v) {
        const int d = d0 + v + 8 * hi;     // C/D layout: M = v + 8*hi, N = r
        partial += acc[tl * 4 + mt][v] * depTa[tl * TPT + d * 16 + r];
      }
    }
    red[wave * 32 + lane] = partial;
    __syncthreads();
    if (tid < 16) {
      float s = bn;
#pragma unroll
      for (int w = 0; w < 8; ++w) s += red[w * 32 + tid] + red[w * 32 + 16 + tid];
      out[(size_t)(b * TT + t0 + tl * 16 + tid) * NLAB + n] = s;
    }
    __syncthreads();   // red reused next tile
  }
}

// ---------------------------------------------------------------------------
extern "C" void kernel_launch(void* const* d_in, const int* in_sizes, int n_in,
                              void* d_out, int out_size, void* d_ws, size_t ws_size,
                              hipStream_t stream) {
  const float* dep   = (const float*)d_in[0];
  const float* head  = (const float*)d_in[1];
  const int*   hidx  = (const int*)d_in[2];
  // d_in[3] = mask: unused by the reference computation
  const float* Wdep  = (const float*)d_in[4];
  const float* bdep  = (const float*)d_in[5];
  const float* Whead = (const float*)d_in[6];
  const float* bhead = (const float*)d_in[7];
  const float* W     = (const float*)d_in[8];
  const float* bias  = (const float*)d_in[9];
  float* out = (float*)d_out;
  float* ws  = (float*)d_ws;   // [0, 4MB): dep_label, [4MB, 8MB): head_label

  const size_t lds1 = (size_t)(16 * DD) * sizeof(float);                 // 64 KB
  const size_t lds2 = (size_t)(4 * TPT + 2 * WBUF + 256) * sizeof(float); // ~273 KB

  proj_label_kernel<<<dim3(ROWS / 16, 4, 2), 128, lds1, stream>>>(
      dep, head, Wdep, bdep, Whead, bhead, ws);
  biaffine_kernel<<<dim3(ROWS / 64, NLAB), 256, lds2, stream>>>(
      ws, hidx, W, bias, out);
}